// HyperConv_64244120814021
// MI455X (gfx1250) — compile-verified
//
#include <hip/hip_runtime.h>
#include <stdint.h>

#define EMB   100
#define QV    25          // float4 chunks per embedding row
#define TILE  256         // nnz staged per LDS tile
#define TPB   4           // tiles processed per block (consecutive)
#define BLOCK 256         // 8 waves (wave32)

// ---------------- CDNA5 async-copy helpers (probe-confirmed present) --------------
#if __has_builtin(__builtin_amdgcn_global_load_async_to_lds_b32)
#define HAVE_ASYNC_LDS 1
typedef __attribute__((address_space(1))) int gas_i32;   // global (device) int*
typedef __attribute__((address_space(3))) int lds_i32;   // LDS int*
#else
#define HAVE_ASYNC_LDS 0
#endif

__device__ __forceinline__ void wait_asynccnt_le3() {
#if __has_builtin(__builtin_amdgcn_s_wait_asynccnt)
  __builtin_amdgcn_s_wait_asynccnt(3);
#else
  asm volatile("s_wait_asynccnt 3" ::: "memory");
#endif
}
__device__ __forceinline__ void wait_asynccnt_le0() {
#if __has_builtin(__builtin_amdgcn_s_wait_asynccnt)
  __builtin_amdgcn_s_wait_asynccnt(0);
#else
  asm volatile("s_wait_asynccnt 0" ::: "memory");
#endif
}

// ---------------- COO SpMM scatter: y[rows[e],:] += vals[e] * x[cols[e],:] --------
// Block owns TPB consecutive 256-nnz tiles; (vals,rows,cols) staged into LDS with
// 2-deep async double buffering (ASYNCcnt). One wave per nnz; lanes 0..24 each
// handle one float4 of the 100-wide row: b128 gather + 4x global_atomic_add_f32.
__global__ __launch_bounds__(BLOCK) void spmm_scatter_kernel(
    const float* __restrict__ vals, const int* __restrict__ rows,
    const int* __restrict__ cols, const float* __restrict__ x,
    float* __restrict__ y, int nnz)
{
  __shared__ float s_val[2][TILE];
  __shared__ int   s_row[2][TILE];
  __shared__ int   s_col[2][TILE];

  const int t    = threadIdx.x;
  const int wave = t >> 5;
  const int lane = t & 31;

  const int ntiles = (nnz + TILE - 1) / TILE;
  const int tile0  = (int)blockIdx.x * TPB;
  int nt = ntiles - tile0;
  if (nt <= 0) return;
  if (nt > TPB) nt = TPB;

  auto stage = [&](int buf, int base) {
    int idx = base + t;
    if (idx >= nnz) idx = nnz - 1;          // clamp: keep addresses valid, data unused
#if HAVE_ASYNC_LDS
    __builtin_amdgcn_global_load_async_to_lds_b32(
        (gas_i32*)(vals + idx), (lds_i32*)&s_val[buf][t], 0, 0);
    __builtin_amdgcn_global_load_async_to_lds_b32(
        (gas_i32*)(rows + idx), (lds_i32*)&s_row[buf][t], 0, 0);
    __builtin_amdgcn_global_load_async_to_lds_b32(
        (gas_i32*)(cols + idx), (lds_i32*)&s_col[buf][t], 0, 0);
#else
    s_val[buf][t] = vals[idx];
    s_row[buf][t] = rows[idx];
    s_col[buf][t] = cols[idx];
#endif
  };

  stage(0, tile0 * TILE);                                  // prologue: tile 0 in flight

  for (int i = 0; i < nt; ++i) {
    const int buf = i & 1;
    if (i + 1 < nt) {
      stage(buf ^ 1, (tile0 + i + 1) * TILE);              // prefetch next tile
      wait_asynccnt_le3();                                 // tile i's 3 async ops landed
    } else {
      wait_asynccnt_le0();
    }
    __syncthreads();                                       // all waves' stages visible

    const int base = (tile0 + i) * TILE;
    #pragma unroll 4
    for (int k = 0; k < TILE / (BLOCK / 32); ++k) {        // 32 nnz per wave
      const int j = wave * (TILE / (BLOCK / 32)) + k;
      if (base + j < nnz) {
        const float v = s_val[buf][j];                     // wave-uniform LDS broadcast
        const int   r = s_row[buf][j];
        const int   c = s_col[buf][j];
        if (lane < QV) {
          const float4 xv = ((const float4*)(x + c * EMB))[lane];
          float* yp = y + r * EMB + lane * 4;              // 32-bit offsets (<= 40 MB)
          __hip_atomic_fetch_add(yp + 0, v * xv.x, __ATOMIC_RELAXED, __HIP_MEMORY_SCOPE_AGENT);
          __hip_atomic_fetch_add(yp + 1, v * xv.y, __ATOMIC_RELAXED, __HIP_MEMORY_SCOPE_AGENT);
          __hip_atomic_fetch_add(yp + 2, v * xv.z, __ATOMIC_RELAXED, __HIP_MEMORY_SCOPE_AGENT);
          __hip_atomic_fetch_add(yp + 3, v * xv.w, __ATOMIC_RELAXED, __HIP_MEMORY_SCOPE_AGENT);
        }
      }
    }
    __syncthreads();                                       // done reading buf before reuse
  }
}

// ---------------- total = x = embedding ; xz = 0 ----------------------------------
__global__ __launch_bounds__(256) void init_kernel(
    const float4* __restrict__ emb, float4* __restrict__ total,
    float4* __restrict__ x, float4* __restrict__ xz, long long n4)
{
  long long i = (long long)blockIdx.x * blockDim.x + threadIdx.x;
  const long long stride = (long long)gridDim.x * blockDim.x;
  const float4 z = make_float4(0.f, 0.f, 0.f, 0.f);
  for (; i < n4; i += stride) { float4 v = emb[i]; total[i] = v; x[i] = v; xz[i] = z; }
}

// ---------------- total += src ; dead = 0 (zeroes next layer's target) ------------
__global__ __launch_bounds__(256) void acc_add_zero_kernel(
    float4* __restrict__ total, const float4* __restrict__ src,
    float4* __restrict__ dead, long long n4)
{
  long long i = (long long)blockIdx.x * blockDim.x + threadIdx.x;
  const long long stride = (long long)gridDim.x * blockDim.x;
  const float4 z = make_float4(0.f, 0.f, 0.f, 0.f);
  for (; i < n4; i += stride) {
    float4 a = total[i]; const float4 b = src[i];
    a.x += b.x; a.y += b.y; a.z += b.z; a.w += b.w;
    total[i] = a;
    dead[i] = z;
  }
}

// ---------------- out[b,:] = user_all[user[b],:] ----------------------------------
__global__ __launch_bounds__(256) void gather_users_kernel(
    const float4* __restrict__ user_all, const int* __restrict__ user,
    float4* __restrict__ out, int batch)
{
  const int i = (int)blockIdx.x * blockDim.x + threadIdx.x;
  if (i >= batch * QV) return;
  const int b = i / QV, q = i % QV;
  out[(long long)b * QV + q] = user_all[(long long)user[b] * QV + q];
}

// ---------------------------------------------------------------------------------
extern "C" void kernel_launch(void* const* d_in, const int* in_sizes, int n_in,
                              void* d_out, int out_size, void* d_ws, size_t ws_size,
                              hipStream_t stream) {
  (void)n_in; (void)out_size; (void)ws_size;
  const float* adj_vals = (const float*)d_in[0];
  const float* u_vals   = (const float*)d_in[1];
  const float* embed    = (const float*)d_in[2];
  /* d_in[3] user_embedding: shape-only in reference */
  const int* adj_rows   = (const int*)d_in[4];
  const int* adj_cols   = (const int*)d_in[5];
  const int* u_rows     = (const int*)d_in[6];
  const int* u_cols     = (const int*)d_in[7];
  const int* user       = (const int*)d_in[8];

  const int nnz_a = in_sizes[0];
  const int nnz_u = in_sizes[1];
  const long long n_items_emb = in_sizes[2];     // N_ITEMS*EMB
  const int batch = in_sizes[8];

  float* total    = (float*)d_out;               // first N_ITEMS*EMB elems
  float* out_user = (float*)d_out + n_items_emb; // then BATCH*EMB elems

  char* ws = (char*)d_ws;
  float* xA = (float*)ws;                                     // 40 MB
  float* xB = (float*)(ws + sizeof(float) * n_items_emb);     // 40 MB

  const long long n4 = n_items_emb / 4;
  const int cgrid = (int)((n4 + 255) / 256);

  // total = xA = embedding; xB = 0 (ready as first scatter target)
  init_kernel<<<cgrid, 256, 0, stream>>>(
      (const float4*)embed, (float4*)total, (float4*)xA, (float4*)xB, n4);

  float* xcur = xA; float* xnext = xB;
  const int ntiles_a = (nnz_a + TILE - 1) / TILE;
  const int sgrid_a  = (ntiles_a + TPB - 1) / TPB;
  for (int l = 0; l < 3; ++l) {
    spmm_scatter_kernel<<<sgrid_a, BLOCK, 0, stream>>>(
        adj_vals, adj_rows, adj_cols, xcur, xnext, nnz_a);
    // total += xnext; zero xcur (it becomes the next layer's scatter target)
    acc_add_zero_kernel<<<cgrid, 256, 0, stream>>>(
        (float4*)total, (const float4*)xnext, (float4*)xcur, n4);
    float* tmp = xcur; xcur = xnext; xnext = tmp;
  }

  // After the loop, xnext points at the buffer zeroed by the last acc_add_zero:
  // use it (first N_USERS*EMB floats) as the pre-zeroed user aggregation scratch.
  float* user_all = xnext;
  const int ntiles_u = (nnz_u + TILE - 1) / TILE;
  const int sgrid_u  = (ntiles_u + TPB - 1) / TPB;
  spmm_scatter_kernel<<<sgrid_u, BLOCK, 0, stream>>>(
      u_vals, u_rows, u_cols, total, user_all, nnz_u);

  const int ggrid = (batch * QV + 255) / 256;
  gather_users_kernel<<<ggrid, 256, 0, stream>>>(
      (const float4*)user_all, user, (float4*)out_user, batch);
}